// WTAMapFusion_28235115004070
// MI455X (gfx1250) — compile-verified
//
#include <hip/hip_runtime.h>

#define GH 400
#define GW 400
#define HW (GH*GW)
#define LH 31
#define LW 51
#define NLOC (LH*LW)          // 1581
#define PH 14
#define PW 36
#define NPOL (PH*PW)          // 504
#define WIN 48                // LDS scatter window (points span <=31 cells)
#define BIGF 1e30f
#define GRESF 0.08f
#define LRESF 0.04f
#define DCOEF 6.25f           // 1/(2*GRES)

typedef float v2f __attribute__((ext_vector_type(2)));
typedef float v8f __attribute__((ext_vector_type(8)));

// ---- order-preserving float<->uint encodings for LDS integer atomics ----
__device__ __forceinline__ unsigned fenc(float f) {          // monotone for all floats
    unsigned u = __float_as_uint(f);
    return (u & 0x80000000u) ? ~u : (u | 0x80000000u);
}
__device__ __forceinline__ float fdec(unsigned u) {
    u = (u & 0x80000000u) ? (u & 0x7FFFFFFFu) : ~u;
    return __uint_as_float(u);
}

// Branchless stencil coefficients (used once at init to fill LDS matrices).
// Dx[k][j]: dhdx[i][j] = sum_k E[i][k] * Dx[k][j]
__device__ __forceinline__ float dx_coef(int k, int j) {
    int jp = min(j + 1, PW - 1);
    int jm = max(j - 1, 0);
    float v = ((k == jp) ? DCOEF : 0.0f) - ((k == jm) ? DCOEF : 0.0f);
    return (j < PW) ? v : 0.0f;
}
// Gy[i][k]: dhdy[i][j] = sum_k Gy[i][k] * E[k][j]
__device__ __forceinline__ float gy_coef(int i, int k) {
    int ip = min(i + 1, PH - 1);
    int im = max(i - 1, 0);
    float v = ((k == ip) ? DCOEF : 0.0f) - ((k == im) ? DCOEF : 0.0f);
    return (i < PH && k < PH) ? v : 0.0f;
}

__global__ void __launch_bounds__(256)
wta_map_fusion_kernel(const float* __restrict__ elev_local,
                      const float* __restrict__ log_var_local,
                      const float* __restrict__ poses,
                      const float* __restrict__ xi,
                      const float* __restrict__ global_elev,
                      const float* __restrict__ global_var,
                      float* __restrict__ out)
{
    __shared__ unsigned s_cmin[WIN * WIN];   // winner variance, positive-float bits
    __shared__ unsigned s_cand[WIN * WIN];   // winner elevation, flip-encoded
    __shared__ float    s_e[16 * 48];        // policy elev samples, zero-padded (WMMA A/B)
    __shared__ float    s_dx[36 * 48];       // Dx stencil matrix, rows k, cols j (padded)
    __shared__ float    s_gy[16 * 16];       // Gy stencil matrix, rows i, cols k (padded)

    const int b   = blockIdx.x;
    const int tid = threadIdx.x;

    const float px = poses[b * 3 + 0];
    const float py = poses[b * 3 + 1];
    const float th = poses[b * 3 + 2];
    const float c  = cosf(th);
    const float s  = sinf(th);

    // Window centered on the transformed local-grid center (1.0, 0.0)
    const float wcx = c * 1.0f + px;
    const float wcy = s * 1.0f + py;
    int gcj = (int)floorf(wcx / GRESF + (float)(GW / 2));
    int gci = (int)floorf(wcy / GRESF + (float)(GH / 2));
    int base_j = min(max(gcj - WIN / 2, 0), GW - WIN);
    int base_i = min(max(gci - WIN / 2, 0), GH - WIN);

    const float* gelev = global_elev + (size_t)b * HW;
    const float* gvar  = global_var  + (size_t)b * HW;

    // ---------------- init LDS (scatter buffers + constant stencil matrices) ----------------
    for (int i = tid; i < WIN * WIN; i += 256) {
        s_cmin[i] = __float_as_uint(BIGF);   // positive float -> uint order OK
        s_cand[i] = fenc(-BIGF);
    }
    for (int i = tid; i < 16 * 48; i += 256) s_e[i] = 0.0f;
    for (int i = tid; i < 36 * 48; i += 256) {
        int k = i / 48, j = i - k * 48;
        s_dx[i] = dx_coef(k, j);
    }
    if (tid < 16 * 16) {
        int i = tid >> 4, k = tid & 15;
        s_gy[tid] = gy_coef(i, k);
    }
    __syncthreads();

    // ---------------- pass 1: scatter-min of winner variance ----------------
    for (int i = tid; i < NLOC; i += 256) {
        int r  = i / LW, cc = i - r * LW;
        float lx = ((float)cc - (float)(LW - 1) * 0.5f) * LRESF + 1.0f;
        float ly = ((float)r  - (float)(LH - 1) * 0.5f) * LRESF;
        float xw = c * lx - s * ly + px;
        float yw = s * lx + c * ly + py;
        int gj = min(max((int)floorf(xw / GRESF + (float)(GW / 2)), 0), GW - 1);
        int gi = min(max((int)floorf(yw / GRESF + (float)(GH / 2)), 0), GH - 1);
        float prior = gvar[gi * GW + gj];
        float var_t = expf(log_var_local[(size_t)b * NLOC + i]);
        float s_eff = fmaxf(var_t, 0.5f * prior);
        bool valid  = (s_eff < 1.5f * prior) || (s_eff < 0.04f);
        float pn = 1.0f / (s_eff + 1e-8f);
        float pp = 1.0f / (prior + 1e-8f);
        float p_win = pn / (pn + pp);
        bool wins = valid && (xi[(size_t)b * NLOC + i] < p_win);
        float mvar = wins ? s_eff : BIGF;
        int li = gi - base_i, lj = gj - base_j;
        if (li >= 0 && li < WIN && lj >= 0 && lj < WIN)
            atomicMin(&s_cmin[li * WIN + lj], __float_as_uint(mvar));
    }
    __syncthreads();

    // ---------------- pass 2: scatter-max of elevation among min-winners ----------------
    for (int i = tid; i < NLOC; i += 256) {
        int r  = i / LW, cc = i - r * LW;
        float lx = ((float)cc - (float)(LW - 1) * 0.5f) * LRESF + 1.0f;
        float ly = ((float)r  - (float)(LH - 1) * 0.5f) * LRESF;
        float xw = c * lx - s * ly + px;
        float yw = s * lx + c * ly + py;
        int gj = min(max((int)floorf(xw / GRESF + (float)(GW / 2)), 0), GW - 1);
        int gi = min(max((int)floorf(yw / GRESF + (float)(GH / 2)), 0), GH - 1);
        float prior = gvar[gi * GW + gj];
        float var_t = expf(log_var_local[(size_t)b * NLOC + i]);
        float s_eff = fmaxf(var_t, 0.5f * prior);
        bool valid  = (s_eff < 1.5f * prior) || (s_eff < 0.04f);
        float pn = 1.0f / (s_eff + 1e-8f);
        float pp = 1.0f / (prior + 1e-8f);
        float p_win = pn / (pn + pp);
        bool wins = valid && (xi[(size_t)b * NLOC + i] < p_win);
        float mvar = wins ? s_eff : BIGF;
        int li = gi - base_i, lj = gj - base_j;
        if (wins && li >= 0 && li < WIN && lj >= 0 && lj < WIN) {
            float cm = __uint_as_float(s_cmin[li * WIN + lj]);
            if (mvar <= cm)
                atomicMax(&s_cand[li * WIN + lj],
                          fenc(elev_local[(size_t)b * NLOC + i]));
        }
    }
    __syncthreads();

    // ---------------- policy-grid bilinear sampling of updated map ----------------
    for (int p = tid; p < NPOL; p += 256) {
        int r  = p / PW, cc = p - r * PW;
        float lx = ((float)cc - (float)(PW - 1) * 0.5f) * GRESF + 0.6f;
        float ly = ((float)r  - (float)(PH - 1) * 0.5f) * GRESF;
        float xw = c * lx - s * ly + px;
        float yw = s * lx + c * ly + py;
        float nx = xw / GRESF / (float)(GW / 2);
        float ny = yw / GRESF / (float)(GH / 2);
        float xp = fminf(fmaxf((nx + 1.0f) * 0.5f * (float)(GW - 1), 0.0f), (float)(GW - 1));
        float yp = fminf(fmaxf((ny + 1.0f) * 0.5f * (float)(GH - 1), 0.0f), (float)(GH - 1));
        int x0 = (int)floorf(xp), y0 = (int)floorf(yp);
        int x1 = min(x0 + 1, GW - 1), y1 = min(y0 + 1, GH - 1);
        float wx = xp - (float)x0, wy = yp - (float)y0;

        float ev[4], vv[4];
        int iys[4] = {y0, y0, y1, y1};
        int ixs[4] = {x0, x1, x0, x1};
        #pragma unroll
        for (int q = 0; q < 4; ++q) {
            int iy = iys[q], ix = ixs[q];
            int li = iy - base_i, lj = ix - base_j;
            bool done = false;
            if (li >= 0 && li < WIN && lj >= 0 && lj < WIN) {
                float cm = __uint_as_float(s_cmin[li * WIN + lj]);
                if (cm < BIGF) {
                    ev[q] = fdec(s_cand[li * WIN + lj]);
                    vv[q] = cm;
                    done = true;
                }
            }
            if (!done) {
                int lin = iy * GW + ix;
                ev[q] = gelev[lin];
                vv[q] = gvar[lin];
            }
        }
        float w00 = (1.0f - wx) * (1.0f - wy), w01 = wx * (1.0f - wy);
        float w10 = (1.0f - wx) * wy,          w11 = wx * wy;
        float es = ev[0]*w00 + ev[1]*w01 + ev[2]*w10 + ev[3]*w11;
        float vs = vv[0]*w00 + vv[1]*w01 + vv[2]*w10 + vv[3]*w11;

        out[((size_t)b * 4 + 0) * NPOL + p] = es;   // elev channel
        out[((size_t)b * 4 + 3) * NPOL + p] = vs;   // var channel
        s_e[r * 48 + cc] = es;                      // staged for WMMA gradients
    }
    __syncthreads();

    // ---------------- gradients via v_wmma_f32_16x16x4_f32 ----------------
    // DX = E(16x48) * Dx(36x36 padded)  -> -dhdx (channel 1)
    // DY = Gy(14x14 padded) * E(16x48)  -> -dhdy (channel 2)
    // ISA 7.12.2 f32 layouts: A 16x4 (M=lane&15, K pair split by lane half across 2 VGPRs);
    // B 4x16 (N=lane&15, K rows {0,2}/{1,3} split by lane half); D vgpr r -> rows r, r+8.
    const int wave = tid >> 5;
    const int lane = tid & 31;
    const int half = lane >> 4;     // 0: lanes 0-15, 1: lanes 16-31
    const int l15  = lane & 15;

    if (wave < 3) {                         // DX output tile: N in [16*wave, 16*wave+16)
        v8f acc = {0.f,0.f,0.f,0.f,0.f,0.f,0.f,0.f};
        const int tn = wave * 16;
        #pragma unroll
        for (int k = 0; k < 9; ++k) {       // K = 36 = 9 * 4
            int kk = k * 4 + half * 2;
            v2f a; a.x = s_e[l15 * 48 + kk];          a.y = s_e[l15 * 48 + kk + 1];
            v2f bm; bm.x = s_dx[kk * 48 + tn + l15];  bm.y = s_dx[(kk + 1) * 48 + tn + l15];
            acc = __builtin_amdgcn_wmma_f32_16x16x4_f32(
                      false, a, false, bm, (short)0, acc, false, false);
        }
        #pragma unroll
        for (int rr = 0; rr < 8; ++rr) {
            int M = half * 8 + rr, N = tn + l15;
            if (M < PH && N < PW)
                out[(((size_t)blockIdx.x * 4 + 1) * PH + M) * PW + N] = -acc[rr];
        }
    } else if (wave < 6) {                  // DY output tile: N in [16*(wave-3), ...)
        v8f acc = {0.f,0.f,0.f,0.f,0.f,0.f,0.f,0.f};
        const int tn = (wave - 3) * 16;
        #pragma unroll
        for (int k = 0; k < 4; ++k) {       // K = 14 padded to 16 = 4 * 4
            int kk = k * 4 + half * 2;
            v2f a; a.x = s_gy[l15 * 16 + kk];         a.y = s_gy[l15 * 16 + kk + 1];
            v2f bm; bm.x = s_e[kk * 48 + tn + l15];   bm.y = s_e[(kk + 1) * 48 + tn + l15];
            acc = __builtin_amdgcn_wmma_f32_16x16x4_f32(
                      false, a, false, bm, (short)0, acc, false, false);
        }
        #pragma unroll
        for (int rr = 0; rr < 8; ++rr) {
            int M = half * 8 + rr, N = tn + l15;
            if (M < PH && N < PW)
                out[(((size_t)blockIdx.x * 4 + 2) * PH + M) * PW + N] = -acc[rr];
        }
    }
}

extern "C" void kernel_launch(void* const* d_in, const int* in_sizes, int n_in,
                              void* d_out, int out_size, void* d_ws, size_t ws_size,
                              hipStream_t stream) {
    (void)n_in; (void)out_size; (void)d_ws; (void)ws_size;
    const float* elev_local    = (const float*)d_in[0];
    const float* log_var_local = (const float*)d_in[1];
    const float* poses         = (const float*)d_in[2];
    const float* xi            = (const float*)d_in[3];
    const float* global_elev   = (const float*)d_in[4];
    const float* global_var    = (const float*)d_in[5];
    float* out = (float*)d_out;
    int B = in_sizes[2] / 3;   // poses is (B, 3)
    wta_map_fusion_kernel<<<dim3(B), dim3(256), 0, stream>>>(
        elev_local, log_var_local, poses, xi, global_elev, global_var, out);
}